// CenterNet_20109036880548
// MI455X (gfx1250) — compile-verified
//
#include <hip/hip_runtime.h>
#include <math.h>

// Problem constants (from setup_inputs: B=16, C=80, H=W=128, K=100)
#define Bb 16
#define Cc 80
#define Hh 128
#define Ww 128
#define HWv 16384
#define Kk 100

// ---------------- gfx1250 async global->LDS path ----------------
#if defined(__has_builtin)
#if __has_builtin(__builtin_amdgcn_global_load_async_to_lds_b128)
#define USE_ASYNC_LDS 1
#endif
#if __has_builtin(__builtin_amdgcn_tensor_load_to_lds)
#define USE_TDM 1
#endif
#endif
#ifndef USE_ASYNC_LDS
#define USE_ASYNC_LDS 0
#endif
#ifndef USE_TDM
#define USE_TDM 0
#endif

typedef int v4i_t __attribute__((vector_size(16)));
typedef __attribute__((address_space(1))) v4i_t* g_v4i_ptr;
typedef __attribute__((address_space(3))) v4i_t* l_v4i_ptr;

typedef unsigned int v4u __attribute__((ext_vector_type(4)));
typedef int v8i __attribute__((ext_vector_type(8)));
typedef int v4i __attribute__((ext_vector_type(4)));

__device__ __forceinline__ void copy16_to_lds(const float* g, float* l) {
#if USE_ASYNC_LDS
  // GLOBAL_LOAD_ASYNC_TO_LDS_B128: per-lane 16B DMA, tracked by ASYNCcnt
  __builtin_amdgcn_global_load_async_to_lds_b128(
      (g_v4i_ptr)(g), (l_v4i_ptr)(l), 0, 0);
#else
  const float4 v = *(const float4*)g;
  *(float4*)l = v;
#endif
}

__device__ __forceinline__ void wait_async_lds() {
#if USE_ASYNC_LDS
  asm volatile("s_wait_asynccnt 0" ::: "memory");
#endif
}

__device__ __forceinline__ void wait_tensor() {
#if defined(__has_builtin) && __has_builtin(__builtin_amdgcn_s_wait_tensorcnt)
  __builtin_amdgcn_s_wait_tensorcnt(0);
#else
  asm volatile("s_wait_tensorcnt 0x0" ::: "memory");
#endif
}

#define NEG_INF (-__builtin_inff())

// comparator: higher key wins; ties -> lower index (matches lax.top_k stability)
__device__ __forceinline__ void take_better(float& bk, int& bi, float ok, int oi) {
  if (ok > bk || (ok == bk && oi < bi)) { bk = ok; bi = oi; }
}

// ---------------- Kernel A: per-(b,c) sigmoid+NMS+top-100 ----------------
// One block per class map. cls map (64KB) staged into LDS once — preferably by
// a single TDM tensor_load_to_lds DMA (TENSORcnt) — then 3x3 local-max mask on
// raw logits (sigmoid is monotone) and 100 iterative argmax passes from LDS.
__global__ __launch_bounds__(256) void centernet_class_topk(
    const float* __restrict__ cls, float* __restrict__ out_sc, int* __restrict__ out_ind) {
  __shared__ float s_raw[HWv];   // raw logits
  __shared__ float s_key[HWv];   // masked logits (NEG_INF if not local max)
  __shared__ float s_rk[256];
  __shared__ int   s_ri[256];

  const int tid = threadIdx.x;
  const int map = blockIdx.x;                       // b*C + c
  const float* src = cls + (size_t)map * HWv;

#if USE_TDM
  if (tid < 32) {  // wave 0 issues one TDM DMA for the whole 64KB tile
    const unsigned lds_base =
        (unsigned)(unsigned long long)(__attribute__((address_space(3))) char*)s_raw;
    const unsigned long long ga = (unsigned long long)src;
    // D# group 0: count=1 | lds_addr | global_addr[56:0] | type=2
    v4u g0 = { 1u, lds_base, (unsigned)ga,
               (unsigned)((ga >> 32) & 0x01ffffffu) | 0x80000000u };
    // D# group 1: data_size=4B; tensor 16384x1; tile 16384x1; stride 16384
    v8i g1 = { 0x00020000,   // workgroup_mask=0, data_size=2 (4 bytes)
               0x40000000,   // tensor_dim0[15:0]=16384 in bits 63:48
               0x00010000,   // tensor_dim1=1 in bits 111:80
               0x40000000,   // tile_dim0=16384 in bits 127:112
               0x00000001,   // tile_dim1=1, tile_dim2=0
               0x00004000,   // tensor_dim0_stride=16384 (low bits)
               0x40000000,   // tensor_dim1_stride[15:0]=16384 in bits 223:208
               0x00000000 }; // tensor_dim1_stride high bits
    v4i gz4 = { 0, 0, 0, 0 };                    // groups 2/3 unused (<=2D tensor)
    v8i gz8 = { 0, 0, 0, 0, 0, 0, 0, 0 };        // extra group (clang-23 6-arg form)
    __builtin_amdgcn_tensor_load_to_lds(g0, g1, gz4, gz4, gz8, 0);
    wait_tensor();
  }
#else
  // stream 64KB map -> LDS via per-lane async DMA (16 x b128 per thread)
  for (int c4 = tid; c4 < HWv / 4; c4 += 256)
    copy16_to_lds(src + 4 * c4, &s_raw[4 * c4]);
  wait_async_lds();
#endif
  __syncthreads();

  // 3x3 max-pool mask (SAME padding == ignore out-of-range neighbors)
  for (int i = 0; i < 64; ++i) {
    const int p = tid + (i << 8);
    const int y = p >> 7, x = p & (Ww - 1);
    const float v = s_raw[p];
    float m = v;
    if (y > 0 && y < Hh - 1 && x > 0 && x < Ww - 1) {
      m = fmaxf(m, s_raw[p - Ww - 1]); m = fmaxf(m, s_raw[p - Ww]); m = fmaxf(m, s_raw[p - Ww + 1]);
      m = fmaxf(m, s_raw[p - 1]);                                    m = fmaxf(m, s_raw[p + 1]);
      m = fmaxf(m, s_raw[p + Ww - 1]); m = fmaxf(m, s_raw[p + Ww]); m = fmaxf(m, s_raw[p + Ww + 1]);
    } else {
      for (int dy = -1; dy <= 1; ++dy)
        for (int dx = -1; dx <= 1; ++dx) {
          const int yy = y + dy, xx = x + dx;
          if (yy >= 0 && yy < Hh && xx >= 0 && xx < Ww)
            m = fmaxf(m, s_raw[yy * Ww + xx]);
        }
    }
    s_key[p] = (v >= m) ? v : NEG_INF;
  }
  __syncthreads();

  float* osc = out_sc + (size_t)map * Kk;
  int*   oin = out_ind + (size_t)map * Kk;

  for (int k = 0; k < Kk; ++k) {
    // per-thread scan: 64 entries, bank-conflict-free stride 256
    float bk = NEG_INF; int bi = 0x7fffffff;
    for (int i = 0; i < 64; ++i) {
      const int p = tid + (i << 8);
      take_better(bk, bi, s_key[p], p);
    }
    s_rk[tid] = bk; s_ri[tid] = bi;
    __syncthreads();
    if (tid < 32) {                       // wave 0 (wave32) reduces 256 -> 1
      bk = s_rk[tid]; bi = s_ri[tid];
      for (int j = 1; j < 8; ++j)
        take_better(bk, bi, s_rk[tid + 32 * j], s_ri[tid + 32 * j]);
      for (int m = 16; m >= 1; m >>= 1) {
        const float ok = __shfl_xor(bk, m, 32);
        const int   oi = __shfl_xor(bi, m, 32);
        take_better(bk, bi, ok, oi);
      }
      if (tid == 0) {
        osc[k] = 1.0f / (1.0f + __expf(-bk));   // sigmoid(-inf) -> 0 (masked)
        oin[k] = bi;
        s_key[bi] = NEG_INF;                    // remove winner
      }
    }
    __syncthreads();
  }
}

// ---------------- Kernel B: per-batch global top-100 + gather + bbox ----------------
__global__ __launch_bounds__(256) void centernet_global_topk(
    const float* __restrict__ reg, const float* __restrict__ wh,
    const float* __restrict__ ws_sc, const int* __restrict__ ws_ind,
    float* __restrict__ out) {
  __shared__ float s_sc[Cc * Kk];     // 8000 candidate scores
  __shared__ int   s_in[Cc * Kk];     // 8000 candidate spatial indices
  __shared__ float s_rk[256];
  __shared__ int   s_ri[256];
  __shared__ float s_selS[Kk];
  __shared__ int   s_selJ[Kk];

  const int tid = threadIdx.x;
  const int b = blockIdx.x;
  const float* sc = ws_sc + (size_t)b * Cc * Kk;
  const int*   in = ws_ind + (size_t)b * Cc * Kk;

  for (int c4 = tid; c4 < (Cc * Kk) / 4; c4 += 256) {
    copy16_to_lds(sc + 4 * c4, &s_sc[4 * c4]);
    copy16_to_lds((const float*)(in + 4 * c4), &((float*)s_in)[4 * c4]);
  }
  wait_async_lds();
  __syncthreads();

  for (int k = 0; k < Kk; ++k) {
    float bk = NEG_INF; int bi = 0x7fffffff;
    for (int j = tid; j < Cc * Kk; j += 256)
      take_better(bk, bi, s_sc[j], j);
    s_rk[tid] = bk; s_ri[tid] = bi;
    __syncthreads();
    if (tid < 32) {
      bk = s_rk[tid]; bi = s_ri[tid];
      for (int j = 1; j < 8; ++j)
        take_better(bk, bi, s_rk[tid + 32 * j], s_ri[tid + 32 * j]);
      for (int m = 16; m >= 1; m >>= 1) {
        const float ok = __shfl_xor(bk, m, 32);
        const int   oi = __shfl_xor(bi, m, 32);
        take_better(bk, bi, ok, oi);
      }
      if (tid == 0) { s_selS[k] = bk; s_selJ[k] = bi; s_sc[bi] = NEG_INF; }
    }
    __syncthreads();
  }

  // gather reg/wh and emit outputs (100 parallel lanes)
  if (tid < Kk) {
    const int k = tid;
    const int j = s_selJ[k];
    const int c = j / Kk;               // class id
    const int ind = s_in[j];            // flat spatial index
    float ys = (float)(ind >> 7);
    float xs = (float)(ind & (Ww - 1));
    const size_t base = ((size_t)b * 2) * HWv;
    const float r0 = reg[base + ind];
    const float r1 = reg[base + HWv + ind];
    const float w0 = wh[base + ind];
    const float w1 = wh[base + HWv + ind];
    xs += r0; ys += r1;
    const float hw2 = w0 * 0.5f, hh2 = w1 * 0.5f;
    const int ob = (b * Kk + k) * 4;
    out[ob + 0] = xs - hw2;
    out[ob + 1] = ys - hh2;
    out[ob + 2] = xs + hw2;
    out[ob + 3] = ys + hh2;
    out[Bb * Kk * 4 + b * Kk + k] = s_selS[k];
    out[Bb * Kk * 4 + Bb * Kk + b * Kk + k] = (float)c;
  }
}

extern "C" void kernel_launch(void* const* d_in, const int* in_sizes, int n_in,
                              void* d_out, int out_size, void* d_ws, size_t ws_size,
                              hipStream_t stream) {
  (void)in_sizes; (void)n_in; (void)out_size; (void)ws_size;
  const float* cls = (const float*)d_in[0];
  const float* reg = (const float*)d_in[1];
  const float* wh  = (const float*)d_in[2];
  // workspace: per-class topk scores [B*C*K] f32, then indices [B*C*K] i32 (~1MB)
  float* ws_sc = (float*)d_ws;
  int*   ws_in = (int*)((char*)d_ws + (size_t)Bb * Cc * Kk * sizeof(float));

  centernet_class_topk<<<Bb * Cc, 256, 0, stream>>>(cls, ws_sc, ws_in);
  centernet_global_topk<<<Bb, 256, 0, stream>>>(reg, wh, ws_sc, ws_in, (float*)d_out);
}